// NR_GraphAttention_52699248722020
// MI455X (gfx1250) — compile-verified
//
#include <hip/hip_runtime.h>
#include <math.h>

typedef __attribute__((ext_vector_type(2))) float v2f;
typedef __attribute__((ext_vector_type(8))) float v8f;

#define D 128

// ---------------------------------------------------------------- K0: tanh(features) -> out[:,0:128]
__global__ void k_tanh_in(const float* __restrict__ feat, float* __restrict__ out,
                          int N, int ostride) {
  int idx = blockIdx.x * blockDim.x + threadIdx.x;
  if (idx < N * D) {
    int n = idx >> 7, k = idx & 127;
    out[(long)n * ostride + k] = tanhf(feat[idx]);
  }
}

// ---------------------------------------------------------------- K1: rel_hat + per-relation attention
// one wave per relation: normalize rel_emb row, dot with each attn kernel
__global__ void k_relhat(const float* __restrict__ rel_emb, const float* __restrict__ attn,
                         float* __restrict__ relhat, float* __restrict__ attrel,
                         int R, int depth) {
  int lane = threadIdx.x & 31;
  int wave = threadIdx.x >> 5;
  int r = blockIdx.x * (blockDim.x >> 5) + wave;
  if (r >= R) return;
  float4 v = ((const float4*)(rel_emb + (long)r * D))[lane];
  float ss = v.x * v.x + v.y * v.y + v.z * v.z + v.w * v.w;
  for (int off = 16; off; off >>= 1) ss += __shfl_xor(ss, off, 32);
  float inv = 1.0f / fmaxf(sqrtf(ss), 1e-12f);
  float4 h = make_float4(v.x * inv, v.y * inv, v.z * inv, v.w * inv);
  ((float4*)(relhat + (long)r * D))[lane] = h;
  for (int l = 0; l < depth; ++l) {
    float4 a = ((const float4*)(attn + (long)l * D))[lane];
    float d = h.x * a.x + h.y * a.y + h.z * a.z + h.w * a.w;
    for (int off = 16; off; off >>= 1) d += __shfl_xor(d, off, 32);
    if (lane == 0) attrel[(long)l * R + r] = d;
  }
}

// ---------------------------------------------------------------- K2: CSR row_ptr from sorted rows
__global__ void k_rowptr(const int* __restrict__ rows, int* __restrict__ rowptr,
                         int N, int E) {
  int n = blockIdx.x * blockDim.x + threadIdx.x;
  if (n > N) return;
  if (n == N) { rowptr[N] = E; return; }
  int lo = 0, hi = E;
  while (lo < hi) {
    int mid = (lo + hi) >> 1;
    if (rows[mid] < n) lo = mid + 1; else hi = mid;
  }
  rowptr[n] = lo;
}

// ---------------------------------------------------------------- K3: one GAT layer, wave-per-row
// featbase: layer input (tanh'd), stride = ostride; outbase: layer output slab, stride = ostride
__global__ void __launch_bounds__(256)
k_layer(const float* __restrict__ featbase, float* __restrict__ outbase,
        const float* __restrict__ relhat, const float* __restrict__ attrel_l,
        const int* __restrict__ adjcol, const int* __restrict__ relid,
        const float* __restrict__ rval, const int* __restrict__ rowptr,
        int N, int ostride) {
  int lane = threadIdx.x & 31;
  int wave = threadIdx.x >> 5;
  int n = blockIdx.x * (blockDim.x >> 5) + wave;
  if (n >= N) return;

  int begin = rowptr[n], end = rowptr[n + 1];
  float4 rowacc = make_float4(0.f, 0.f, 0.f, 0.f);

  if (begin < end) {
    // ---- softmax stats over this row's edges (att depends only on relation id)
    float amax = -__builtin_inff();
    for (int e = begin + lane; e < end; e += 32) {
      float rv = rval[e];
      float att = (rv > 0.f) ? attrel_l[relid[e]] : 0.f;
      amax = fmaxf(amax, att);
    }
    for (int off = 16; off; off >>= 1) amax = fmaxf(amax, __shfl_xor(amax, off, 32));
    float ssum = 0.f;
    for (int e = begin + lane; e < end; e += 32) {
      float rv = rval[e];
      float att = (rv > 0.f) ? attrel_l[relid[e]] : 0.f;
      ssum += __expf(att - amax);
    }
    for (int off = 16; off; off >>= 1) ssum += __shfl_xor(ssum, off, 32);
    float sinv = 1.f / ssum;

    int m = lane & 15;              // edge slot within 16-edge tile
    int khalf = (lane >> 4) << 1;   // A/B fragment K-half: 0 or 2
    int sel = (lane < 8) ? lane : (lane >= 24 ? lane - 24 : -1);  // diag owner map

    // ---- 16-edge tiles: WMMA computes all 16 Householder dots as diag of 16x16
    for (int t = begin; t < end; t += 16) {
      int valid = end - t; if (valid > 16) valid = 16;
      int e = (m < valid) ? (t + m) : begin;   // pad with a safe edge, weight 0
      int col = adjcol[e];
      int rel = relid[e];
      float rv = rval[e];
      float att = (rv > 0.f) ? attrel_l[rel] : 0.f;
      float w = (m < valid) ? __expf(att - amax) * sinv : 0.f;
      float sref = (rv > 0.f) ? 2.f : 0.f;     // kills reflection when tri_rel == 0

      const float* xrow = featbase + (long)col * ostride;
      const float* hrow = relhat + (long)rel * D;

      v8f acc = {};
      #pragma unroll
      for (int kb = 0; kb < D; kb += 4) {
        // A[m][k]: lane L holds row m=L%16, K = kb + khalf + {0,1}  (16x4 f32 layout)
        v2f a = *(const v2f*)(xrow + kb + khalf);
        // B[k][n]: column n = rel_hat of edge n; same lane addressing, different base
        v2f b = *(const v2f*)(hrow + kb + khalf);
        acc = __builtin_amdgcn_wmma_f32_16x16x4_f32(
            false, a, false, b, (short)0, acc, false, false);
      }

      // diagonal extraction: lanes 0-7 -> acc[lane] (M=N=lane); lanes 24-31 -> acc[lane-24]
      float dsel = 0.f;
      #pragma unroll
      for (int j = 0; j < 8; ++j) dsel = (sel == j) ? acc[j] : dsel;

      // ---- weighted Householder accumulate; lane owns output dims [4*lane, 4*lane+4)
      for (int m2 = 0; m2 < valid; ++m2) {
        int srcd = (m2 < 8) ? m2 : (m2 + 16);
        float dm = __shfl(dsel, srcd, 32);
        float wm = __shfl(w, m2, 32);
        float cm = __shfl(sref, m2, 32) * dm;   // 2*dot (or 0)
        int colm = __shfl(col, m2, 32);
        int relm = __shfl(rel, m2, 32);
        float4 x4 = *(const float4*)(featbase + (long)colm * ostride + 4 * lane);
        float4 h4 = *(const float4*)(relhat + (long)relm * D + 4 * lane);
        rowacc.x += wm * (x4.x - cm * h4.x);
        rowacc.y += wm * (x4.y - cm * h4.y);
        rowacc.z += wm * (x4.z - cm * h4.z);
        rowacc.w += wm * (x4.w - cm * h4.w);
      }
    }
  }

  // fused tanh + atomic-free store (this wave exclusively owns row n)
  float4 o;
  o.x = tanhf(rowacc.x); o.y = tanhf(rowacc.y);
  o.z = tanhf(rowacc.z); o.w = tanhf(rowacc.w);
  *(float4*)(outbase + (long)n * ostride + 4 * lane) = o;
}

// ----------------------------------------------------------------
extern "C" void kernel_launch(void* const* d_in, const int* in_sizes, int n_in,
                              void* d_out, int out_size, void* d_ws, size_t ws_size,
                              hipStream_t stream) {
  const float* features = (const float*)d_in[0];
  const float* rel_emb  = (const float*)d_in[1];
  const int*   adj      = (const int*)d_in[2];   // [2,E]: rows then cols
  const int*   r_index  = (const int*)d_in[3];   // [2,E]: arange then relation ids
  const float* r_val    = (const float*)d_in[4];
  const float* attn     = (const float*)d_in[8]; // [depth, 128]

  int N = in_sizes[0] / D;
  int R = in_sizes[1] / D;
  int E = in_sizes[4];
  int depth = in_sizes[8] / D;
  int ostride = D * (depth + 1);
  float* out = (float*)d_out;

  // workspace: rel_hat [R*128] f32 | att_rel [depth*R] f32 | row_ptr [N+1] i32
  char* ws = (char*)d_ws;
  float* relhat = (float*)ws;
  float* attrel = (float*)(ws + (size_t)R * D * 4);
  int*   rowptr = (int*)(ws + (size_t)R * D * 4 + (size_t)depth * R * 4);

  k_tanh_in<<<(N * D + 255) / 256, 256, 0, stream>>>(features, out, N, ostride);
  k_relhat<<<(R + 7) / 8, 256, 0, stream>>>(rel_emb, attn, relhat, attrel, R, depth);
  k_rowptr<<<(N + 256) / 256, 256, 0, stream>>>(adj, rowptr, N, E);

  for (int l = 0; l < depth; ++l) {
    k_layer<<<(N + 7) / 8, 256, 0, stream>>>(
        out + (size_t)D * l,            // layer input slab (already tanh'd)
        out + (size_t)D * (l + 1),      // layer output slab
        relhat, attrel + (size_t)l * R,
        adj + E, r_index + E, r_val, rowptr,
        N, ostride);
  }
}